// HGTLayer_27565100105880
// MI455X (gfx1250) — compile-verified
//
#include <hip/hip_runtime.h>

// ---------------- problem constants (match reference) ----------------
#define N_NODES 32768
#define DEG     8
#define E_EDGES (N_NODES * DEG)          // 262144
#define LG_EDGES (E_EDGES * DEG)         // 2097152
#define D       128
#define H       8
#define DK      16
#define NEG_SLOPE 0.01f
#define NEG_FLT_MAX (-3.402823466e38f)

typedef __attribute__((ext_vector_type(2))) float v2f;
typedef __attribute__((ext_vector_type(8))) float v8f;

// =====================================================================
// fp32 WMMA GEMM:  Y[rows x 128] = leaky?( [X0 | X1] @ W_t + b_t )
//   - KTOT = K0 + K1 (128 or 256); X1 null when K1 == 0
//   - per-type weights: type = rowBase / rowsPerType, W = [KTOT x 128]
//     per type (row-major, reference einsum 'tni,tio->tno')
//   - block = 256 threads = 8 waves; wave w owns columns [16w,16w+16)
//   - block computes MT row-tiles (MT*16 rows); each wave keeps MT
//     independent v8f accumulators so consecutive v_wmma_f32_16x16x4_f32
//     never RAW-chain on the same D registers, and each B fragment is
//     reused MT times.
//   - A tile staged in LDS (padded stride: 16 lanes -> distinct banks)
//   - V_WMMA_F32_16X16X4_F32 fragment layouts (wave32):
//     A (16x4): lane<16 -> row=lane,   k(v0),k+1(v1); lane>=16 -> row=lane-16, k+2,k+3
//     B (4x16): lane<16 -> col=lane,   k(v0),k+1(v1); lane>=16 -> col=lane-16, k+2,k+3
//     C/D     : vgpr r : lanes 0-15 -> M=r ; lanes 16-31 -> M=8+r ; N=lane&15
// =====================================================================
template <int KTOT, int MT, bool LEAKY>
__global__ __launch_bounds__(256) void gemm_wmma_f32(
    const float* __restrict__ X0, const float* __restrict__ X1,
    const float* __restrict__ W, const float* __restrict__ bias,
    float* __restrict__ Y, int rowsPerType, int K0)
{
    constexpr int LDA  = KTOT + 4;               // floats; 16B-aligned rows
    constexpr int ROWS = MT * 16;
    __shared__ float As[ROWS * LDA];

    const int rowBase = blockIdx.x * ROWS;
    const int type    = rowBase / rowsPerType;
    const float* Wt   = W + (size_t)type * KTOT * D;
    const float* bt   = bias + (size_t)type * D;
    const int K1      = KTOT - K0;

    // cooperative A-tile stage: ROWS x KTOT floats, float4 granularity
    const int K4 = KTOT / 4;
    for (int idx = threadIdx.x; idx < ROWS * K4; idx += 256) {
        const int r  = idx / K4;
        const int c4 = idx - r * K4;
        float4 v;
        if (c4 < K0 / 4) v = ((const float4*)(X0 + (size_t)(rowBase + r) * K0))[c4];
        else             v = ((const float4*)(X1 + (size_t)(rowBase + r) * K1))[c4 - K0 / 4];
        *(float4*)&As[r * LDA + 4 * c4] = v;
    }
    __syncthreads();

    const int lane    = threadIdx.x & 31;
    const int wave    = threadIdx.x >> 5;        // 0..7 -> 16-col tile
    const int cBase   = wave * 16;
    const int halfSel = lane >> 4;               // 0 | 1
    const int l15     = lane & 15;

    v8f acc[MT];
#pragma unroll
    for (int t = 0; t < MT; ++t) acc[t] = {};

    const float* Wc = Wt + cBase + l15;          // this lane's output column

#pragma unroll 2
    for (int k = 0; k < KTOT; k += 4) {
        const int ka = k + 2 * halfSel;
        v2f b;
        b.x = Wc[(size_t)ka * D];
        b.y = Wc[(size_t)(ka + 1) * D];
#pragma unroll
        for (int t = 0; t < MT; ++t) {
            v2f a;
            a.x = As[(t * 16 + l15) * LDA + ka];
            a.y = As[(t * 16 + l15) * LDA + ka + 1];
            acc[t] = __builtin_amdgcn_wmma_f32_16x16x4_f32(
                         /*neg_a=*/false, a, /*neg_b=*/false, b,
                         /*c_mod=*/(short)0, acc[t],
                         /*reuse_a=*/false, /*reuse_b=*/false);
        }
    }

    const float bv = bt[cBase + l15];
#pragma unroll
    for (int t = 0; t < MT; ++t) {
#pragma unroll
        for (int r = 0; r < 8; ++r) {
            float v = acc[t][r] + bv;
            if (LEAKY) v = (v > 0.0f) ? v : NEG_SLOPE * v;
            const int m = t * 16 + r + 8 * halfSel;
            Y[(size_t)(rowBase + m) * D + cBase + l15] = v;
        }
    }
}

// =====================================================================
// small helpers
// =====================================================================
__global__ void fill_f32(float* __restrict__ p, float v, int n) {
    const int i = blockIdx.x * blockDim.x + threadIdx.x;
    if (i < n) p[i] = v;
}

__device__ __forceinline__ void atomicMaxF32(float* addr, float val) {
    unsigned int* ua = (unsigned int*)addr;
    unsigned int  old = *ua;
    while (__uint_as_float(old) < val) {
        const unsigned int assumed = old;
        old = atomicCAS(ua, assumed, __float_as_uint(val));
        if (old == assumed) break;
    }
}

// xe = h_e + h_n[src]   (float4 vectorized)
__global__ void edge_input_kernel(const float* __restrict__ h_e,
                                  const float* __restrict__ h_n,
                                  const int* __restrict__ src,
                                  float* __restrict__ xe)
{
    const int i = blockIdx.x * blockDim.x + threadIdx.x;   // E * D/4
    if (i >= E_EDGES * (D / 4)) return;
    const int e  = i >> 5;              // / (D/4)
    const int c4 = i & 31;
    const float4 a = ((const float4*)h_e)[i];
    const float4 b = ((const float4*)(h_n + (size_t)src[e] * D))[c4];
    ((float4*)xe)[i] = make_float4(a.x + b.x, a.y + b.y, a.z + b.z, a.w + b.w);
}

// =====================================================================
// attention 1: graph edges -> node messages
// =====================================================================
__global__ void att1_logits(const float* __restrict__ Qn, const float* __restrict__ Ke,
                            const int* __restrict__ dst,
                            float* __restrict__ logits, float* __restrict__ segmax)
{
    const int i = blockIdx.x * blockDim.x + threadIdx.x;   // E*H
    if (i >= E_EDGES * H) return;
    const int e = i >> 3, h = i & 7;
    const int d = dst[e];
    const float4* q = (const float4*)(Qn + (size_t)d * D + h * DK);
    const float4* k = (const float4*)(Ke + (size_t)e * D + h * DK);
    float s = 0.0f;
#pragma unroll
    for (int j = 0; j < 4; ++j) {
        const float4 a = q[j], b = k[j];
        s += a.x * b.x + a.y * b.y + a.z * b.z + a.w * b.w;
    }
    s *= 0.25f;                                            // 1/sqrt(DK)
    logits[i] = s;
    atomicMaxF32(&segmax[d * H + h], s);
}

__global__ void att1_accum(const float* __restrict__ Ve, const int* __restrict__ dst,
                           const float* __restrict__ logits, const float* __restrict__ segmax,
                           float* __restrict__ segsum, float* __restrict__ m_n)
{
    const int i = blockIdx.x * blockDim.x + threadIdx.x;   // E*H
    if (i >= E_EDGES * H) return;
    const int e = i >> 3, h = i & 7;
    const int d = dst[e];
    const float w = __expf(logits[i] - segmax[d * H + h]);
    atomicAdd(&segsum[d * H + h], w);
    const float* v   = Ve + (size_t)e * D + h * DK;
    float*       out = m_n + (size_t)d * D + h * DK;
#pragma unroll
    for (int j = 0; j < DK; ++j) atomicAdd(&out[j], w * v[j]);
}

__global__ void att_norm(float* __restrict__ m, const float* __restrict__ segsum, int rows)
{
    const int i = blockIdx.x * blockDim.x + threadIdx.x;   // rows*D
    if (i >= rows * D) return;
    const int r = i >> 7;                                  // / D
    const int h = (i & 127) >> 4;                          // head
    const float s = segsum[r * H + h];
    m[i] = (s > 0.0f) ? m[i] / s : 0.0f;                   // empty segment -> 0 (matches ref)
}

// =====================================================================
// attention 2: line-graph (LG edges) + self K/V-from-node terms (E edges)
//   i < LG : q-seg = lg_dst[i], K/V from Ke/Ve[lg_src[i]]
//   i >= LG: q-seg = e = i-LG , K/V from Kn/Vn[src[e]]
// =====================================================================
__global__ void att2_logits(const float* __restrict__ Qe,
                            const float* __restrict__ Ke, const float* __restrict__ Kn,
                            const int* __restrict__ lg_src, const int* __restrict__ lg_dst,
                            const int* __restrict__ src,
                            float* __restrict__ logits, float* __restrict__ segmax)
{
    const int i = blockIdx.x * blockDim.x + threadIdx.x;   // (LG+E)*H
    if (i >= (LG_EDGES + E_EDGES) * H) return;
    const int idx = i >> 3, h = i & 7;
    int qi; const float* kb;
    if (idx < LG_EDGES) { qi = lg_dst[idx]; kb = Ke + (size_t)lg_src[idx] * D; }
    else { const int e = idx - LG_EDGES; qi = e; kb = Kn + (size_t)src[e] * D; }
    const float4* q = (const float4*)(Qe + (size_t)qi * D + h * DK);
    const float4* k = (const float4*)(kb + h * DK);
    float s = 0.0f;
#pragma unroll
    for (int j = 0; j < 4; ++j) {
        const float4 a = q[j], b = k[j];
        s += a.x * b.x + a.y * b.y + a.z * b.z + a.w * b.w;
    }
    s *= 0.25f;
    logits[i] = s;
    atomicMaxF32(&segmax[qi * H + h], s);
}

__global__ void att2_accum(const float* __restrict__ Ve, const float* __restrict__ Vn,
                           const int* __restrict__ lg_src, const int* __restrict__ lg_dst,
                           const int* __restrict__ src,
                           const float* __restrict__ logits, const float* __restrict__ segmax,
                           float* __restrict__ segsum, float* __restrict__ m_e)
{
    const int i = blockIdx.x * blockDim.x + threadIdx.x;   // (LG+E)*H
    if (i >= (LG_EDGES + E_EDGES) * H) return;
    const int idx = i >> 3, h = i & 7;
    int qi; const float* vb;
    if (idx < LG_EDGES) { qi = lg_dst[idx]; vb = Ve + (size_t)lg_src[idx] * D; }
    else { const int e = idx - LG_EDGES; qi = e; vb = Vn + (size_t)src[e] * D; }
    const float w = __expf(logits[i] - segmax[qi * H + h]);
    atomicAdd(&segsum[qi * H + h], w);
    const float* v   = vb + h * DK;
    float*       out = m_e + (size_t)qi * D + h * DK;
#pragma unroll
    for (int j = 0; j < DK; ++j) atomicAdd(&out[j], w * v[j]);
}

// =====================================================================
// host-side orchestration
// =====================================================================
extern "C" void kernel_launch(void* const* d_in, const int* in_sizes, int n_in,
                              void* d_out, int out_size, void* d_ws, size_t ws_size,
                              hipStream_t stream)
{
    const float* h_n    = (const float*)d_in[0];
    const float* h_e    = (const float*)d_in[1];
    const int*   src    = (const int*)d_in[2];
    const int*   dst    = (const int*)d_in[3];
    const int*   lg_src = (const int*)d_in[4];
    const int*   lg_dst = (const int*)d_in[5];
    const float* n_q_W = (const float*)d_in[6];   const float* n_q_b = (const float*)d_in[7];
    const float* n_k_W = (const float*)d_in[8];   const float* n_k_b = (const float*)d_in[9];
    const float* n_v_W = (const float*)d_in[10];  const float* n_v_b = (const float*)d_in[11];
    const float* e_q_W = (const float*)d_in[12];  const float* e_q_b = (const float*)d_in[13];
    const float* e_k_W = (const float*)d_in[14];  const float* e_k_b = (const float*)d_in[15];
    const float* e_v_W = (const float*)d_in[16];  const float* e_v_b = (const float*)d_in[17];
    const float* tm_W  = (const float*)d_in[18];  const float* tm_b  = (const float*)d_in[19];
    const float* n_lin_W = (const float*)d_in[20]; const float* n_lin_b = (const float*)d_in[21];
    const float* Wnd_W = (const float*)d_in[22];  const float* Wnd_b = (const float*)d_in[23];
    const float* Wed_W = (const float*)d_in[24];  const float* Wed_b = (const float*)d_in[25];

    const float* tmn_W = tm_W;                 const float* tmn_b = tm_b;          // node types
    const float* tme_W = tm_W + 2 * D * D;     const float* tme_b = tm_b + 2 * D;  // edge types

    // ---- workspace layout (floats) ----
    const size_t fN = (size_t)N_NODES * D;
    const size_t fE = (size_t)E_EDGES * D;
    float* ws      = (float*)d_ws;
    float* tmp     = ws;  ws += fE;                 // chained-GEMM intermediate (max rows = E)
    float* Qn      = ws;  ws += fN;
    float* Kn      = ws;  ws += fN;
    float* Vn      = ws;  ws += fN;
    float* xe_me   = ws;  ws += fE;                 // xe, then reused as m_e
    float* Qe      = ws;  ws += fE;                 // later reused for u_e
    float* Ke      = ws;  ws += fE;
    float* Ve      = ws;  ws += fE;
    float* m_n     = ws;  ws += fN;
    float* segmax  = ws;  ws += (size_t)E_EDGES * H;
    float* segsum  = ws;  ws += (size_t)E_EDGES * H;
    float* logits  = ws;  ws += (size_t)(LG_EDGES + E_EDGES) * H;

    float* out_n = (float*)d_out;
    float* out_e = out_n + fN;

    // K=128 path: 4 row-tiles per block (64 rows);  K=256 path: 2 row-tiles (32 rows)
    const int GN128 = N_NODES / 64, GE128 = E_EDGES / 64;
    const int GN256 = N_NODES / 32, GE256 = E_EDGES / 32;

    // ---- phase 1: node Q/K/V (two chained per-type GEMMs each) ----
    gemm_wmma_f32<128,4,false><<<GN128, 256, 0, stream>>>(h_n, nullptr, n_q_W, n_q_b, tmp, N_NODES/2, 128);
    gemm_wmma_f32<128,4,false><<<GN128, 256, 0, stream>>>(tmp, nullptr, tmn_W, tmn_b, Qn, N_NODES/2, 128);
    gemm_wmma_f32<128,4,false><<<GN128, 256, 0, stream>>>(h_n, nullptr, n_k_W, n_k_b, tmp, N_NODES/2, 128);
    gemm_wmma_f32<128,4,false><<<GN128, 256, 0, stream>>>(tmp, nullptr, tmn_W, tmn_b, Kn, N_NODES/2, 128);
    gemm_wmma_f32<128,4,false><<<GN128, 256, 0, stream>>>(h_n, nullptr, n_v_W, n_v_b, tmp, N_NODES/2, 128);
    gemm_wmma_f32<128,4,false><<<GN128, 256, 0, stream>>>(tmp, nullptr, tmn_W, tmn_b, Vn, N_NODES/2, 128);

    // ---- phase 2: edge features + edge Q/K/V ----
    edge_input_kernel<<<(E_EDGES * (D/4) + 255)/256, 256, 0, stream>>>(h_e, h_n, src, xe_me);
    gemm_wmma_f32<128,4,false><<<GE128, 256, 0, stream>>>(xe_me, nullptr, e_q_W, e_q_b, tmp, E_EDGES/2, 128);
    gemm_wmma_f32<128,4,false><<<GE128, 256, 0, stream>>>(tmp, nullptr, tme_W, tme_b, Qe, E_EDGES/2, 128);
    gemm_wmma_f32<128,4,false><<<GE128, 256, 0, stream>>>(xe_me, nullptr, e_k_W, e_k_b, tmp, E_EDGES/2, 128);
    gemm_wmma_f32<128,4,false><<<GE128, 256, 0, stream>>>(tmp, nullptr, tme_W, tme_b, Ke, E_EDGES/2, 128);
    gemm_wmma_f32<128,4,false><<<GE128, 256, 0, stream>>>(xe_me, nullptr, e_v_W, e_v_b, tmp, E_EDGES/2, 128);
    gemm_wmma_f32<128,4,false><<<GE128, 256, 0, stream>>>(tmp, nullptr, tme_W, tme_b, Ve, E_EDGES/2, 128);

    // ---- phase 3: attention 1 (edges -> nodes), 3-pass segment softmax ----
    fill_f32<<<(N_NODES*H + 255)/256, 256, 0, stream>>>(segmax, NEG_FLT_MAX, N_NODES*H);
    fill_f32<<<(N_NODES*H + 255)/256, 256, 0, stream>>>(segsum, 0.0f, N_NODES*H);
    fill_f32<<<((int)fN + 255)/256,  256, 0, stream>>>(m_n, 0.0f, (int)fN);
    att1_logits<<<(E_EDGES*H + 255)/256, 256, 0, stream>>>(Qn, Ke, dst, logits, segmax);
    att1_accum <<<(E_EDGES*H + 255)/256, 256, 0, stream>>>(Ve, dst, logits, segmax, segsum, m_n);
    att_norm   <<<((int)fN + 255)/256,   256, 0, stream>>>(m_n, segsum, N_NODES);

    // ---- phase 4: attention 2 (line-graph + node self terms -> edges) ----
    fill_f32<<<(E_EDGES*H + 255)/256, 256, 0, stream>>>(segmax, NEG_FLT_MAX, E_EDGES*H);
    fill_f32<<<(E_EDGES*H + 255)/256, 256, 0, stream>>>(segsum, 0.0f, E_EDGES*H);
    fill_f32<<<((int)fE + 255)/256,   256, 0, stream>>>(xe_me, 0.0f, (int)fE);   // m_e
    {
        const int tot = (LG_EDGES + E_EDGES) * H;
        att2_logits<<<(tot + 255)/256, 256, 0, stream>>>(Qe, Ke, Kn, lg_src, lg_dst, src, logits, segmax);
        att2_accum <<<(tot + 255)/256, 256, 0, stream>>>(Ve, Vn, lg_src, lg_dst, src, logits, segmax, segsum, xe_me);
        att_norm   <<<((int)fE + 255)/256, 256, 0, stream>>>(xe_me, segsum, E_EDGES);
    }

    // ---- phase 5: outputs ----
    // u_n = m_n @ n_lin + b (single type) ; h_n_out = leaky([h_n | u_n] @ Wnd + b)
    gemm_wmma_f32<128,4,false><<<GN128, 256, 0, stream>>>(m_n, nullptr, n_lin_W, n_lin_b, Qn, N_NODES, 128);
    gemm_wmma_f32<256,2,true ><<<GN256, 256, 0, stream>>>(h_n, Qn, Wnd_W, Wnd_b, out_n, N_NODES/2, 128);
    // u_e = m_e @ n_lin + b ; h_e_out = leaky([h_e | u_e] @ Wed + b)
    gemm_wmma_f32<128,4,false><<<GE128, 256, 0, stream>>>(xe_me, nullptr, n_lin_W, n_lin_b, Qe, E_EDGES, 128);
    gemm_wmma_f32<256,2,true ><<<GE256, 256, 0, stream>>>(h_e, Qe, Wed_W, Wed_b, out_e, E_EDGES/2, 128);
}